// RandomProjectionQuantizer_28475633172775
// MI455X (gfx1250) — compile-verified
//
#include <hip/hip_runtime.h>
#include <stdint.h>

// Problem constants from the reference
#define B_ 4
#define L_ 512
#define D_ 1024
#define Q_ 256
#define C_ 256

#define SLICES_PER_BLOCK 16              // (b,l) slices written per block
#define NBLOCKS ((B_ * L_) / SLICES_PER_BLOCK)   // 2048/16 = 128

// ---------------------------------------------------------------------------
// Fused single-launch kernel.
//
// Math (reference collapses algebraically):
//   labels[b,l,c] = argmin_q( LN(t)[b,l,c] - LN(cb)[c,q] )
//                 = argmax_q LN(cb)[c,q]           (argmin arg is const in q)
//                 = argmax_q code_book[c,q]        (LN w/ global stats is a
//                                                   strictly increasing map)
// so the output is the per-row argmax of code_book broadcast over (b,l).
//
// Every block recomputes all 256 row-argmaxes (256 KB of reads, L2-resident
// across all 128 blocks given 192 MB L2) so the whole job is ONE launch —
// at this size launch latency dominates the ~100 ns of HBM traffic.
//
// CDNA5 usage: each wave32 double-buffers its codebook rows through LDS via
// global_load_async_to_lds_b128 (ASYNCcnt), overlapping the next row's DMA
// with the current row's argmax reduction; s_wait_asynccnt 2 drains only the
// landed row (async loads signal done in order). Reduction uses wave32
// shuffle-xor butterflies; the final broadcast is pure b128 store traffic.
// ---------------------------------------------------------------------------
__global__ __launch_bounds__(256) void rpq_fused(const float* __restrict__ code_book,
                                                 float* __restrict__ out) {
    __shared__ __align__(16) float stage[8][2][Q_];  // per-wave double buffer, 16 KB
    __shared__ __align__(16) float slabel[C_];       // the 256 labels, 1 KB

    const int tid  = threadIdx.x;     // 256 threads = 8 wave32
    const int lane = tid & 31;
    const int wave = tid >> 5;

    // Per-lane LDS byte addresses (low 32 bits of a generic LDS pointer are
    // the group-segment offset). Each lane lands 16 B at lds + lane*16, and
    // the offset:512 copy covers the second half of the 1 KB row buffer —
    // INST_OFFSET is added to BOTH the LDS and global addresses (ISA §4.4).
    const unsigned lds_buf0 = (unsigned)(uintptr_t)&stage[wave][0][0] + (unsigned)(lane * 16);
    const unsigned lds_buf1 = (unsigned)(uintptr_t)&stage[wave][1][0] + (unsigned)(lane * 16);

    const int row0 = wave * 32;       // this wave owns rows [row0, row0+32)

    // Prologue: stage row0 into buffer 0.
    {
        unsigned long long g = (unsigned long long)(uintptr_t)(code_book + (size_t)row0 * Q_)
                             + (unsigned long long)(lane * 16);
        asm volatile("global_load_async_to_lds_b128 %0, %1, off"            :: "v"(lds_buf0), "v"(g) : "memory");
        asm volatile("global_load_async_to_lds_b128 %0, %1, off offset:512" :: "v"(lds_buf0), "v"(g) : "memory");
    }

    for (int i = 0; i < 32; ++i) {
        const int row = row0 + i;

        // Issue the NEXT row's async copies into the other buffer, then wait
        // for the CURRENT row (2 oldest async ops) to land. Safe to overwrite
        // the other buffer: its previous contents were consumed by VALU in
        // iteration i-1, which forced the ds_load waits to retire already.
        if (i < 31) {
            const unsigned nxt = (i & 1) ? lds_buf0 : lds_buf1;
            unsigned long long g = (unsigned long long)(uintptr_t)(code_book + (size_t)(row + 1) * Q_)
                                 + (unsigned long long)(lane * 16);
            asm volatile("global_load_async_to_lds_b128 %0, %1, off"            :: "v"(nxt), "v"(g) : "memory");
            asm volatile("global_load_async_to_lds_b128 %0, %1, off offset:512" :: "v"(nxt), "v"(g) : "memory");
            asm volatile("s_wait_asynccnt 0x2" ::: "memory");  // row i landed; row i+1 in flight
        } else {
            asm volatile("s_wait_asynccnt 0x0" ::: "memory");  // drain the last row
        }

        // Argmax of the staged row; lane owns 8 contiguous columns.
        const float4* b4 = (const float4*)&stage[wave][i & 1][0];
        float4 a = b4[lane * 2 + 0];                 // ds_load_b128
        float4 b = b4[lane * 2 + 1];                 // ds_load_b128
        const float v[8] = {a.x, a.y, a.z, a.w, b.x, b.y, b.z, b.w};

        float best = v[0];
        int   bidx = lane * 8;
        #pragma unroll
        for (int k = 1; k < 8; ++k) {
            // strictly-greater => first occurrence of the max in the chunk
            if (v[k] > best) { best = v[k]; bidx = lane * 8 + k; }
        }

        // wave32 butterfly; on value tie the smaller index wins, matching
        // jnp.argmin's first-occurrence semantics exactly.
        #pragma unroll
        for (int off = 16; off > 0; off >>= 1) {
            float ov = __shfl_xor(best, off, 32);
            int   oi = __shfl_xor(bidx, off, 32);
            if (ov > best || (ov == best && oi < bidx)) { best = ov; bidx = oi; }
        }

        if (lane == 0) slabel[row] = (float)bidx;    // ds_store_b32
    }

    __syncthreads();   // all 256 labels visible to all waves

    // Broadcast phase: this block writes slices [blockIdx.x*16, +16) of the
    // 2048 (b,l) slices; each slice is C_=256 floats = 64 float4. 16 KB of
    // pure global_store_b128 traffic per block, 2 MiB total.
    float4*       out4 = (float4*)(out + (size_t)blockIdx.x * SLICES_PER_BLOCK * C_);
    const float4* s4   = (const float4*)slabel;

    #pragma unroll
    for (int k = 0; k < 4; ++k) {
        const int i  = k * 256 + tid;   // 0..1023 float4 slots in this block
        const int c4 = i & 63;          // float4 index within a slice
        out4[i] = s4[c4];
    }
}

// ---------------------------------------------------------------------------
// Launch. Inputs (setup_inputs dict order):
//   d_in[0] = input_values (B,L,D) f32  -- provably unused by the reference
//   d_in[1] = W            (Q,D)   f32  -- provably unused by the reference
//   d_in[2] = code_book    (C,Q)   f32
//   d_in[3] = raw_signal   scalar int
// d_out: B*L*C = 524288 floats. Single kernel launch; no workspace needed.
// ---------------------------------------------------------------------------
extern "C" void kernel_launch(void* const* d_in, const int* in_sizes, int n_in,
                              void* d_out, int out_size, void* d_ws, size_t ws_size,
                              hipStream_t stream) {
    (void)in_sizes; (void)n_in; (void)out_size; (void)d_ws; (void)ws_size;

    const float* code_book = (const float*)d_in[2];
    float*       out       = (float*)d_out;

    rpq_fused<<<dim3(NBLOCKS), dim3(256), 0, stream>>>(code_book, out);
}